// FilterPool_1932735283388
// MI455X (gfx1250) — compile-verified
//
#include <hip/hip_runtime.h>

typedef __attribute__((ext_vector_type(2))) float v2f;
typedef __attribute__((ext_vector_type(8))) float v8f;

namespace {
constexpr int kH = 36, kW = 36, kC = 256, kK = kH * kW;   // 1296
constexpr float kScale = 1.0f / 16.0f;

// Antiderivative of the triangular bilinear kernel, as in the reference.
__device__ __forceinline__ float gfun(float u) {
  u = fminf(fmaxf(u, -1.0f), 1.0f);
  float p = u + 1.0f;
  float q = 1.0f - u;
  return (u < 0.0f) ? 0.5f * p * p : 1.0f - 0.5f * q * q;
}
}  // namespace

// One block = 4 waves x 16 channels = 64 channels of one ROI.
// grid = (C/64, N). Each wave does a 16-channel GEMV over K=1296 via
// chained v_wmma_f32_16x16x4_f32 with the weight vector replicated
// across the 16 B-columns (compute is free; kernel is HBM-bound).
__global__ __launch_bounds__(128) void prroi_wmma_kernel(
    const float* __restrict__ feat,
    const float* __restrict__ bb,
    float* __restrict__ out) {
  __shared__ float s_wx[kW];
  __shared__ float s_wy[kH];
  __shared__ float s_w[kK];

  const int n    = blockIdx.y;
  const int tid  = threadIdx.x;
  const int lane = tid & 31;
  const int wave = tid >> 5;

  // ---- per-ROI separable weights (redundant per block; trivial cost) ----
  const float bx = bb[n * 4 + 0], by = bb[n * 4 + 1];
  const float bw = bb[n * 4 + 2], bh = bb[n * 4 + 3];
  const float x0 = bx * kScale, y0 = by * kScale;
  const float x1 = (bx + bw) * kScale, y1 = (by + bh) * kScale;
  const float binw = x1 - x0, binh = y1 - y0;
  const float area = fmaxf(binw, 0.0f) * fmaxf(binh, 0.0f);
  const float inv  = (area > 0.0f) ? (1.0f / fmaxf(area, 1e-30f)) : 0.0f;

  if (tid < kW) {
    s_wx[tid] = gfun(x1 - (float)tid) - gfun(x0 - (float)tid);
  } else if (tid >= 64 && tid < 64 + kH) {
    const int i = tid - 64;
    s_wy[i] = gfun(y1 - (float)i) - gfun(y0 - (float)i);
  }
  __syncthreads();
  for (int k = tid; k < kK; k += 128) {
    // fold 1/area into the weights so D is the final output
    s_w[k] = s_wy[k / kW] * s_wx[k % kW] * inv;
  }
  __syncthreads();

  // ---- 16-channel GEMV via WMMA ----
  const int c0   = blockIdx.x * 64 + wave * 16;
  const int m    = lane & 15;          // A row (channel) within the tile
  const int koff = (lane >> 4) * 2;    // K sub-offset per ISA 16x4 A layout
  const float* __restrict__ arow =
      feat + ((size_t)n * kC + (size_t)(c0 + m)) * (size_t)kK + koff;

  v8f acc0 = {}, acc1 = {};   // two chains to hide WMMA latency
#pragma unroll 2
  for (int k = 0; k < kK; k += 8) {
    v2f a0 = *(const v2f*)(arow + k);
    v2f a1 = *(const v2f*)(arow + k + 4);
    v2f b0 = { s_w[k + koff],     s_w[k + koff + 1]     };
    v2f b1 = { s_w[k + 4 + koff], s_w[k + 4 + koff + 1] };
    acc0 = __builtin_amdgcn_wmma_f32_16x16x4_f32(
        /*neg_a=*/false, a0, /*neg_b=*/false, b0,
        /*c_mod=*/(short)0, acc0, /*reuse_a=*/false, /*reuse_b=*/false);
    acc1 = __builtin_amdgcn_wmma_f32_16x16x4_f32(
        false, a1, false, b1, (short)0, acc1, false, false);
  }
  v8f acc = acc0 + acc1;

  // D layout: VGPR r, lane 0  -> (M=r,   N=0)
  //           VGPR r, lane 16 -> (M=r+8, N=0); all columns identical anyway.
  float* __restrict__ o = out + n * kC + c0;
  if (lane == 0) {
#pragma unroll
    for (int r = 0; r < 8; ++r) o[r] = acc[r];
  } else if (lane == 16) {
#pragma unroll
    for (int r = 0; r < 8; ++r) o[8 + r] = acc[r];
  }
}

extern "C" void kernel_launch(void* const* d_in, const int* in_sizes, int n_in,
                              void* d_out, int out_size, void* d_ws, size_t ws_size,
                              hipStream_t stream) {
  (void)n_in; (void)d_ws; (void)ws_size; (void)out_size;
  const float* feat = (const float*)d_in[0];
  const float* bb   = (const float*)d_in[1];
  float* out        = (float*)d_out;

  const int N = in_sizes[1] / 4;       // bb is (N, 4)
  dim3 grid(kC / 64, N);               // 4 channel-groups x N ROIs
  dim3 block(128);                     // 4 waves x wave32
  prroi_wmma_kernel<<<grid, block, 0, stream>>>(feat, bb, out);
}